// FakeModule_4337916969126
// MI455X (gfx1250) — compile-verified
//
#include <hip/hip_runtime.h>
#include <math.h>

#define H 4
#define D 32
#define N 256

typedef __attribute__((ext_vector_type(2))) float v2f;
typedef __attribute__((ext_vector_type(8))) float v8f;
typedef __attribute__((ext_vector_type(4))) int v4i;

#if defined(__gfx1250__) &&                                          \
    __has_builtin(__builtin_amdgcn_global_load_async_to_lds_b128) && \
    __has_builtin(__builtin_amdgcn_s_wait_asynccnt)
#define USE_ASYNC_LDS 1
// b128 async copy operands: int4* in global (AS1) and LDS (AS3) address spaces
typedef __attribute__((address_space(1))) v4i* glb_v4i_p;
typedef __attribute__((address_space(3))) v4i* lds_v4i_p;
#else
#define USE_ASYNC_LDS 0
#endif

__device__ __forceinline__ v8f wmma_f32_16x16x4(v2f a, v2f b, v8f c) {
  // 8 args: (neg_a, A, neg_b, B, c_mod, C, reuse_a, reuse_b)
  return __builtin_amdgcn_wmma_f32_16x16x4_f32(false, a, false, b, (short)0, c,
                                               false, false);
}

// ---------------------------------------------------------------------------
// Kernel A: triangle bias  tri[h, j, k] = x[j,k,:] . Wb[h,:] + bb[h]
// [65536 x 32] x [32 x 4] -> too thin for WMMA tiles; memory-bound VALU.
// ---------------------------------------------------------------------------
__global__ void tri_bias_kernel(const float* __restrict__ x,
                                const float* __restrict__ Wb,
                                const float* __restrict__ bb,
                                float* __restrict__ tri) {
  int idx = blockIdx.x * blockDim.x + threadIdx.x;  // h*65536 + j*256 + k
  if (idx >= H * N * N) return;
  int h  = idx >> 16;
  int jk = idx & 0xFFFF;
  const float4* xr = (const float4*)(x + (size_t)jk * D);
  const float4* wr = (const float4*)(Wb + h * D);
  float acc = bb[h];
#pragma unroll
  for (int c4 = 0; c4 < D / 4; ++c4) {
    float4 xv = xr[c4], wv = wr[c4];
    acc += xv.x * wv.x + xv.y * wv.y + xv.z * wv.z + xv.w * wv.w;
  }
  tri[idx] = acc;
}

// ---------------------------------------------------------------------------
// Kernel B: fused QKV projection + flash attention for one (head h, row i).
// 512 threads = 16 waves; wave w owns query rows j = 16w .. 16w+15.
// ---------------------------------------------------------------------------
__global__ __launch_bounds__(512, 1) void tri_attn_kernel(
    const float* __restrict__ x, const float* __restrict__ mask,
    const float* __restrict__ Wq, const float* __restrict__ bq,
    const float* __restrict__ Wk, const float* __restrict__ bk,
    const float* __restrict__ Wv, const float* __restrict__ bv,
    const float* __restrict__ tri, float* __restrict__ out) {
  const int i    = blockIdx.x;
  const int h    = blockIdx.y;
  const int tid  = threadIdx.x;
  const int wave = tid >> 5;
  const int lane = tid & 31;
  const int l15  = lane & 15;
  const int hi   = lane >> 4;  // 0 or 1 (which 16-lane half)

  const int SX = 33;  // padded row stride (floats) for 32-wide tiles
  __shared__ float s_x[N * 33];        // x[i] tile; later reused: mask + P bufs
  __shared__ float s_q[N * 33];
  __shared__ float s_k[N * 33];
  __shared__ float s_v[N * 33];
  __shared__ float s_w[3 * 32 * 33];   // Wq_h, Wk_h, Wv_h as [d][c]

  float* s_mask = s_x;                        // 256 floats (after projection)
  float* s_p    = s_x + 256 + wave * 16 * 17; // per-wave 16x16 P tile, stride 17

  // ---- stage x[i] (256x32) and head weight slices into LDS ----
  const float* xi = x + (size_t)i * N * D;
#if USE_ASYNC_LDS
  // Async direct-to-LDS copies (no VGPR round trip); b128 groups of 4 floats
  // stay within a padded row since 4 | 32.  One s_wait_asynccnt covers all.
  for (int t = tid; t < N * (D / 4); t += 512) {
    int r = t >> 3, g = (t & 7) * 4;
    __builtin_amdgcn_global_load_async_to_lds_b128(
        (glb_v4i_p)(xi + r * D + g), (lds_v4i_p)&s_x[r * SX + g], 0, 0);
  }
  for (int t = tid; t < 32 * (32 / 4); t += 512) {
    int d = t >> 3, g = (t & 7) * 4;
    __builtin_amdgcn_global_load_async_to_lds_b128(
        (glb_v4i_p)(Wq + (h * 32 + d) * 32 + g),
        (lds_v4i_p)&s_w[0 * 32 * SX + d * SX + g], 0, 0);
    __builtin_amdgcn_global_load_async_to_lds_b128(
        (glb_v4i_p)(Wk + (h * 32 + d) * 32 + g),
        (lds_v4i_p)&s_w[1 * 32 * SX + d * SX + g], 0, 0);
    __builtin_amdgcn_global_load_async_to_lds_b128(
        (glb_v4i_p)(Wv + (h * 32 + d) * 32 + g),
        (lds_v4i_p)&s_w[2 * 32 * SX + d * SX + g], 0, 0);
  }
  __builtin_amdgcn_s_wait_asynccnt(0);
#else
  for (int t = tid; t < N * D; t += 512) {
    int r = t >> 5, c = t & 31;
    s_x[r * SX + c] = xi[t];
  }
  for (int t = tid; t < 32 * 32; t += 512) {
    int d = t >> 5, c = t & 31;
    s_w[0 * 32 * SX + d * SX + c] = Wq[(h * 32 + d) * 32 + c];
    s_w[1 * 32 * SX + d * SX + c] = Wk[(h * 32 + d) * 32 + c];
    s_w[2 * 32 * SX + d * SX + c] = Wv[(h * 32 + d) * 32 + c];
  }
#endif
  __syncthreads();

  // ---- phase 1: Q/K/V projection for this wave's 16-row block ----
  const int m0 = wave * 16;
  {
    const float* biasv[3] = {bq, bk, bv};
    float* dst[3] = {s_q, s_k, s_v};
#pragma unroll
    for (int which = 0; which < 3; ++which) {
      const float* sw = s_w + which * 32 * SX;
#pragma unroll
      for (int dt = 0; dt < 2; ++dt) {
        const int dcol = dt * 16 + l15;
        const float bv_ = biasv[which][h * 32 + dcol];
        v8f acc = {bv_, bv_, bv_, bv_, bv_, bv_, bv_, bv_};
#pragma unroll
        for (int cc = 0; cc < 32; cc += 4) {
          const int ck = cc + 2 * hi;
          v2f a;  // A: row = m0+l15, K = ck/ck+1
          a.x = s_x[(m0 + l15) * SX + ck];
          a.y = s_x[(m0 + l15) * SX + ck + 1];
          v2f b;  // B[c][d] = W[d][c]: row = ck/ck+1, col = dcol
          b.x = sw[dcol * SX + ck];
          b.y = sw[dcol * SX + ck + 1];
          acc = wmma_f32_16x16x4(a, b, acc);
        }
#pragma unroll
        for (int r = 0; r < 8; ++r)
          dst[which][(m0 + r + 8 * hi) * SX + dcol] = acc[r];
      }
    }
  }
  __syncthreads();  // K/V visible to all waves; s_x region now free

  if (tid < N) s_mask[tid] = mask[i * N + tid];
  __syncthreads();

  // ---- phase 2: flash attention ----
  v2f qf[8];  // Q A-fragments, reused across all key blocks
#pragma unroll
  for (int cc = 0; cc < 32; cc += 4) {
    const int ck = cc + 2 * hi;
    qf[cc >> 2].x = s_q[(m0 + l15) * SX + ck];
    qf[cc >> 2].y = s_q[(m0 + l15) * SX + ck + 1];
  }

  float mrow[8], lrow[8];
#pragma unroll
  for (int r = 0; r < 8; ++r) { mrow[r] = -INFINITY; lrow[r] = 0.0f; }
  v8f o0 = {}, o1 = {};
  const float scale = 0.17677669529663687f;  // 1/sqrt(32)
  const float* trih = tri + (size_t)h * N * N;

  for (int kb = 0; kb < N / 16; ++kb) {
    const int kk0 = kb * 16;

    // Prefetch next key block's tri-bias lines (L2-resident table) so the
    // loads below hit while the QK^T WMMA chain executes.
    if (kb + 1 < N / 16) {
#pragma unroll
      for (int r = 0; r < 8; ++r) {
        const int jrow = m0 + r + 8 * hi;
        __builtin_prefetch(&trih[jrow * N + kk0 + 16 + l15], 0, 0);
      }
    }

    // S = Q @ K^T  (16x16, contract over d=32)
    v8f s = {};
#pragma unroll
    for (int cc = 0; cc < 32; cc += 4) {
      const int ck = cc + 2 * hi;
      v2f b;  // B[d][kk] = K[kk][d]
      b.x = s_k[(kk0 + l15) * SX + ck];
      b.y = s_k[(kk0 + l15) * SX + ck + 1];
      s = wmma_f32_16x16x4(qf[cc >> 2], b, s);
    }

    // bias + mask + online softmax (rows live across a 16-lane half)
    const int kcol = kk0 + l15;
    const float mk = (s_mask[kcol] - 1.0f) * 1e9f;
#pragma unroll
    for (int r = 0; r < 8; ++r) {
      const int jrow = m0 + r + 8 * hi;
      float sv = s[r] * scale + trih[jrow * N + kcol] + mk;
      float rm = sv;
      rm = fmaxf(rm, __shfl_xor(rm, 1, 16));
      rm = fmaxf(rm, __shfl_xor(rm, 2, 16));
      rm = fmaxf(rm, __shfl_xor(rm, 4, 16));
      rm = fmaxf(rm, __shfl_xor(rm, 8, 16));
      const float mnew  = fmaxf(mrow[r], rm);
      const float alpha = __expf(mrow[r] - mnew);
      const float pv    = __expf(sv - mnew);
      float rs = pv;
      rs += __shfl_xor(rs, 1, 16);
      rs += __shfl_xor(rs, 2, 16);
      rs += __shfl_xor(rs, 4, 16);
      rs += __shfl_xor(rs, 8, 16);
      lrow[r] = lrow[r] * alpha + rs;
      mrow[r] = mnew;
      o0[r] *= alpha;
      o1[r] *= alpha;
      s_p[(r + 8 * hi) * 17 + l15] = pv;  // C-layout -> A-layout via LDS
    }

    // O += P @ V  (contract over 16 keys; two 16-wide d tiles)
#pragma unroll
    for (int cc = 0; cc < 16; cc += 4) {
      const int ck = cc + 2 * hi;
      v2f a;  // A: row = local j = l15, K = local kk
      a.x = s_p[l15 * 17 + ck];
      a.y = s_p[l15 * 17 + ck + 1];
      v2f b0, b1;  // B[kk][d]
      b0.x = s_v[(kk0 + ck) * SX + l15];
      b0.y = s_v[(kk0 + ck + 1) * SX + l15];
      b1.x = s_v[(kk0 + ck) * SX + 16 + l15];
      b1.y = s_v[(kk0 + ck + 1) * SX + 16 + l15];
      o0 = wmma_f32_16x16x4(a, b0, o0);
      o1 = wmma_f32_16x16x4(a, b1, o1);
    }
  }

  // ---- epilogue: out[i, j, h, d] = O / l ----
#pragma unroll
  for (int r = 0; r < 8; ++r) {
    const int jrow  = m0 + r + 8 * hi;
    const float inv = 1.0f / lrow[r];
    float* op = out + (((size_t)i * N + jrow) * H + h) * D;
    op[l15]      = o0[r] * inv;
    op[16 + l15] = o1[r] * inv;
  }
}

extern "C" void kernel_launch(void* const* d_in, const int* in_sizes, int n_in,
                              void* d_out, int out_size, void* d_ws,
                              size_t ws_size, hipStream_t stream) {
  (void)in_sizes; (void)n_in; (void)out_size; (void)ws_size;
  const float* x    = (const float*)d_in[0];
  const float* mask = (const float*)d_in[1];
  const float* Wq   = (const float*)d_in[2];
  const float* bq   = (const float*)d_in[3];
  const float* Wk   = (const float*)d_in[4];
  const float* bk   = (const float*)d_in[5];
  const float* Wv   = (const float*)d_in[6];
  const float* bv   = (const float*)d_in[7];
  const float* Wb   = (const float*)d_in[8];
  const float* bb   = (const float*)d_in[9];
  float* out = (float*)d_out;
  float* tri = (float*)d_ws;  // H*N*N floats = 1 MB scratch

  tri_bias_kernel<<<dim3((H * N * N + 255) / 256), 256, 0, stream>>>(x, Wb, bb,
                                                                     tri);
  tri_attn_kernel<<<dim3(N, H), 512, 0, stream>>>(x, mask, Wq, bq, Wk, bk, Wv,
                                                  bv, tri, out);
}